// ESPnetStatistic_80814104642186
// MI455X (gfx1250) — compile-verified
//
#include <hip/hip_runtime.h>

#define BINS   100
#define BLOCK  256
#define WAVES  (BLOCK / 32)
#define VPT    4                    // float4 staged per thread per chunk
#define CHUNK  (BLOCK * VPT)        // 1024 float4 = 16 KB per buffer

// Low 32 bits of a generic pointer to __shared__ data are the wave-relative
// LDS byte address (CDNA5 ISA 10.2, LDS aperture).
__device__ __forceinline__ unsigned lds_byte_addr(const void* p) {
    return (unsigned)(unsigned long long)p;
}

// CDNA5 async copy: global memory -> LDS, 16 bytes per lane, tracked by ASYNCcnt.
__device__ __forceinline__ void async_ld_b128(unsigned lds_addr, const void* gptr) {
    asm volatile("global_load_async_to_lds_b128 %0, %1, off"
                 :
                 : "v"(lds_addr), "v"((unsigned long long)gptr)
                 : "memory");
}

__device__ __forceinline__ void online_update4(float4 v, float& m, float& s) {
    float mx = fmaxf(fmaxf(v.x, v.y), fmaxf(v.z, v.w));
    if (mx > m) { s *= __expf(m - mx); m = mx; }
    s += __expf(v.x - m) + __expf(v.y - m) + __expf(v.z - m) + __expf(v.w - m);
}

// One workgroup per (b,t) row: online softmax denominator over V logits,
// then conf = exp(x[y]-max)/sum (or -1 when the position is padded).
__global__ __launch_bounds__(BLOCK)
void conf_kernel(const float* __restrict__ logits,
                 const int*   __restrict__ ys,
                 const int*   __restrict__ lens,
                 float*       __restrict__ conf,
                 int T, int V)
{
    const int row  = blockIdx.x;
    const int tid  = threadIdx.x;
    const int lane = tid & 31;
    const int wave = tid >> 5;

    const size_t rowbase = (size_t)row * (size_t)V;
    const float4* rowp4  = (const float4*)(logits + rowbase);
    const int nvec       = V >> 2;          // 32000 -> 8000 float4
    const int main_iters = nvec / CHUNK;    // 7 staged chunks

    __shared__ float4 stage[2][CHUNK];      // 32 KB double buffer
    __shared__ float  red_m[WAVES];
    __shared__ float  red_s[WAVES];

    // per-thread LDS byte addresses of its VPT slots in each buffer
    unsigned lds[2][VPT];
#pragma unroll
    for (int k = 0; k < VPT; ++k) {
        lds[0][k] = lds_byte_addr(&stage[0][tid + k * BLOCK]);
        lds[1][k] = lds_byte_addr(&stage[1][tid + k * BLOCK]);
    }

    float m = -3.402823466e38f;
    float s = 0.0f;

    // --- async-staged main loop (double buffered, epilogue peeled) ---
    if (main_iters > 0) {
#pragma unroll
        for (int k = 0; k < VPT; ++k)       // prefetch chunk 0
            async_ld_b128(lds[0][k], rowp4 + (size_t)k * BLOCK + tid);

        for (int it = 0; it < main_iters - 1; ++it) {
            const int cur = it & 1, nxt = cur ^ 1;
            // prior ds reads of `nxt` slots are complete before overwrite
            asm volatile("s_wait_dscnt 0" ::: "memory");
#pragma unroll
            for (int k = 0; k < VPT; ++k)
                async_ld_b128(lds[nxt][k],
                              rowp4 + (size_t)(it + 1) * CHUNK + k * BLOCK + tid);
            asm volatile("s_wait_asynccnt 4" ::: "memory");  // chunk `it` landed

            float4 v0 = stage[cur][tid + 0 * BLOCK];
            float4 v1 = stage[cur][tid + 1 * BLOCK];
            float4 v2 = stage[cur][tid + 2 * BLOCK];
            float4 v3 = stage[cur][tid + 3 * BLOCK];
            // one rescale decision per 16 elements, then 16 exps
            float mx = fmaxf(fmaxf(fmaxf(fmaxf(v0.x, v0.y), fmaxf(v0.z, v0.w)),
                                   fmaxf(fmaxf(v1.x, v1.y), fmaxf(v1.z, v1.w))),
                             fmaxf(fmaxf(fmaxf(v2.x, v2.y), fmaxf(v2.z, v2.w)),
                                   fmaxf(fmaxf(v3.x, v3.y), fmaxf(v3.z, v3.w))));
            if (mx > m) { s *= __expf(m - mx); m = mx; }
            s += __expf(v0.x - m) + __expf(v0.y - m) + __expf(v0.z - m) + __expf(v0.w - m)
               + __expf(v1.x - m) + __expf(v1.y - m) + __expf(v1.z - m) + __expf(v1.w - m)
               + __expf(v2.x - m) + __expf(v2.y - m) + __expf(v2.z - m) + __expf(v2.w - m)
               + __expf(v3.x - m) + __expf(v3.y - m) + __expf(v3.z - m) + __expf(v3.w - m);
        }

        // epilogue: last staged chunk
        asm volatile("s_wait_asynccnt 0" ::: "memory");
        const int cur = (main_iters - 1) & 1;
#pragma unroll
        for (int k = 0; k < VPT; ++k)
            online_update4(stage[cur][tid + k * BLOCK], m, s);
    }

    // --- tail (nvec % CHUNK vectors) via direct b128 loads ---
    for (int idx = main_iters * CHUNK + tid; idx < nvec; idx += BLOCK) {
        float4 v = rowp4[idx];
        online_update4(v, m, s);
    }

    // --- wave32 reduction of (max, sum) pairs ---
#pragma unroll
    for (int off = 16; off > 0; off >>= 1) {
        float mo = __shfl_xor(m, off, 32);
        float so = __shfl_xor(s, off, 32);
        float mn = fmaxf(m, mo);
        s = s * __expf(m - mn) + so * __expf(mo - mn);
        m = mn;
    }
    if (lane == 0) { red_m[wave] = m; red_s[wave] = s; }
    __syncthreads();

    if (tid == 0) {
        m = red_m[0]; s = red_s[0];
#pragma unroll
        for (int w = 1; w < WAVES; ++w) {
            float mo = red_m[w], so = red_s[w];
            float mn = fmaxf(m, mo);
            s = s * __expf(m - mn) + so * __expf(mo - mn);
            m = mn;
        }
        const int y   = ys[row];
        const float xy = logits[rowbase + (size_t)y];
        const int b = row / T;
        const int t = row - b * T;
        const bool valid = t < lens[b];
        conf[row] = valid ? (__expf(xy - m) / s) : -1.0f;
    }
}

// Single workgroup: 100-bin histogram with the reference's exact bounds
// (lo = i/100 by true division, hi = i + 1/100). Writes all of d_out.
__global__ __launch_bounds__(BLOCK)
void hist_kernel(const float* __restrict__ conf,
                 float*       __restrict__ out,
                 int nrows)
{
    __shared__ unsigned cnt[BINS];
    const int tid = threadIdx.x;

    for (int i = tid; i < BINS; i += BLOCK) cnt[i] = 0u;
    __syncthreads();

    for (int j = tid; j < nrows; j += BLOCK) {
        const float c = conf[j];
        if (c > 0.0f) {                       // padded rows (-1) hit no bin
            for (int i = 0; i < BINS; ++i) {
                const float lo = (float)i / 100.0f;
                const float hi = (float)i + 0.01f;
                if (c > lo && c < hi) atomicAdd(&cnt[i], 1u);
            }
        }
    }
    __syncthreads();

    for (int i = tid; i < BINS; i += BLOCK) out[i] = (float)cnt[i];
}

extern "C" void kernel_launch(void* const* d_in, const int* in_sizes, int n_in,
                              void* d_out, int out_size, void* d_ws, size_t ws_size,
                              hipStream_t stream)
{
    const float* logits = (const float*)d_in[0];
    const int*   ys     = (const int*)d_in[1];
    const int*   lens   = (const int*)d_in[2];

    const int BT = in_sizes[1];          // B*T = 8192
    const int B  = in_sizes[2];          // 8
    const int T  = BT / B;               // 1024
    const int V  = in_sizes[0] / BT;     // 32000

    float* conf = (float*)d_ws;          // BT floats of scratch

    conf_kernel<<<BT, BLOCK, 0, stream>>>(logits, ys, lens, conf, T, V);
    hist_kernel<<<1, BLOCK, 0, stream>>>(conf, (float*)d_out, BT);
}